// HIDTA_48146583388801
// MI455X (gfx1250) — compile-verified
//
#include <hip/hip_runtime.h>
#include <hip/hip_bf16.h>
#include <math.h>

// ---------------- problem constants ----------------
#define B_   32
#define N_   500
#define T0_  13
#define E_   2000
#define EH_  300
#define C_   40
#define EPS_ 1e-5f

#define ACT_NONE  0
#define ACT_SIG   1
#define ACT_TANH  2
#define ACT_LEAKY 3
#define ACT_ACC   8   // flag bit: accumulate into D

typedef float v2f __attribute__((ext_vector_type(2)));
typedef float v8f __attribute__((ext_vector_type(8)));
typedef long long i64;

static __device__ __forceinline__ float sigmoid_(float x) {
    return 1.0f / (1.0f + __expf(-x));
}

// D/C tile store: lanes 0-15: n=col0+lane, rows row0+r (r=0..7); lanes 16-31: rows row0+8+r
static __device__ __forceinline__ void st_tile(float* __restrict__ D, int d_rs,
                                               const float* __restrict__ bias,
                                               int M, int Ncol, int row0, int col0,
                                               v8f acc, int act, bool accum) {
    int lane = threadIdx.x & 31;
    int n = col0 + (lane & 15);
    int mb = row0 + ((lane >> 4) << 3);
    if (n >= Ncol) return;
#pragma unroll
    for (int r = 0; r < 8; ++r) {
        int m = mb + r;
        if (m < M) {
            float v = acc[r];
            if (bias) v += bias[m];
            if      (act == ACT_SIG)   v = sigmoid_(v);
            else if (act == ACT_TANH)  v = tanhf(v);
            else if (act == ACT_LEAKY) v = v > 0.0f ? v : 0.01f * v;
            unsigned adr = (unsigned)m * (unsigned)d_rs + (unsigned)n;
            if (accum) v += D[adr];
            D[adr] = v;
        }
    }
}

// ------------- generic batched WMMA GEMM (f32 16x16x4), 32x64 tile per wave -------------
// D[z] = act(A[z] * B[z] + bias) (optionally += D)
// BMODE 0: B addr = k*b_rs + n  (affine)
// BMODE 1: B addr = (k>>1)*b_rs + (k&1)*b_aux + n  (dilated conv, K'=2*C, k0 step 4 keeps k even)
// K must be a multiple of 4. Out-of-range rows/cols are index-clamped on load (results
// land in D rows/cols the store guard skips), so the K-loop is branch- and mask-free.
template <int BMODE>
__global__ __launch_bounds__(128) void gwnet_gemm_wmma(
    const float* __restrict__ A, int a_rs, int a_cs, i64 a_bs,
    const float* __restrict__ Bm, int b_rs, i64 b_bs, int b_aux,
    float* __restrict__ D, int d_rs, i64 d_bs,
    const float* __restrict__ bias,
    int M, int Ncol, int K, int epi)
{
    const int lane = threadIdx.x & 31;
    const int row0 = blockIdx.y * 32;
    const int col0 = (blockIdx.x * 4 + threadIdx.y) * 64;
    if (col0 >= Ncol) return;
    const int z = blockIdx.z;
    const float* __restrict__ Ab = A + (i64)z * a_bs;
    const float* __restrict__ Bb = Bm + (i64)z * b_bs;
    float* __restrict__ Db = D + (i64)z * d_bs;

    const int lr = lane & 15;
    const int lk = (lane >> 4) << 1;   // 0 or 2

    // A fragment offsets (two row tiles), 32-bit, advanced by astep per K-step
    int mA0 = row0 + lr;      if (mA0 >= M) mA0 = M - 1;
    int mA1 = row0 + 16 + lr; if (mA1 >= M) mA1 = M - 1;
    unsigned a0off = (unsigned)mA0 * (unsigned)a_rs + (unsigned)lk * (unsigned)a_cs;
    unsigned a1off = (unsigned)mA1 * (unsigned)a_rs + (unsigned)lk * (unsigned)a_cs;
    const unsigned acs   = (unsigned)a_cs;
    const unsigned astep = 4u * acs;

    // B fragment offsets (four col tiles)
    unsigned boff[4];
    const unsigned bsec  = (BMODE == 0) ? (unsigned)b_rs : (unsigned)b_aux;  // +1 k element
    const unsigned bstep = (BMODE == 0) ? 4u * (unsigned)b_rs : 2u * (unsigned)b_rs;
    const unsigned kterm = (BMODE == 0) ? (unsigned)lk * (unsigned)b_rs
                                        : (unsigned)(lk >> 1) * (unsigned)b_rs;
#pragma unroll
    for (int j = 0; j < 4; ++j) {
        int n = col0 + j * 16 + lr;
        if (n >= Ncol) n = Ncol - 1;
        boff[j] = kterm + (unsigned)n;
    }

    v8f acc0[4], acc1[4];
#pragma unroll
    for (int j = 0; j < 4; ++j) {
        acc0[j] = (v8f){0,0,0,0,0,0,0,0};
        acc1[j] = (v8f){0,0,0,0,0,0,0,0};
    }

    for (int k0 = 0; k0 < K; k0 += 4) {
        v2f a0, a1, bf[4];
        a0.x = Ab[a0off];
        a0.y = Ab[a0off + acs];
        a1.x = Ab[a1off];
        a1.y = Ab[a1off + acs];
#pragma unroll
        for (int j = 0; j < 4; ++j) {
            bf[j].x = Bb[boff[j]];
            bf[j].y = Bb[boff[j] + bsec];
        }
        // prefetch the streaming operand ~8 K-steps ahead (global_prefetch_b8)
        __builtin_prefetch(Bb + boff[0] + 8u * bstep, 0, 1);
        a0off += astep;
        a1off += astep;
#pragma unroll
        for (int j = 0; j < 4; ++j) boff[j] += bstep;
#pragma unroll
        for (int j = 0; j < 4; ++j) {
            acc0[j] = __builtin_amdgcn_wmma_f32_16x16x4_f32(false, a0, false, bf[j], (short)0, acc0[j], false, false);
            acc1[j] = __builtin_amdgcn_wmma_f32_16x16x4_f32(false, a1, false, bf[j], (short)0, acc1[j], false, false);
        }
    }

    const int act = epi & 7;
    const bool accum = (epi & ACT_ACC) != 0;
#pragma unroll
    for (int j = 0; j < 4; ++j) {
        st_tile(Db, d_rs, bias, M, Ncol, row0,      col0 + j * 16, acc0[j], act, accum);
        st_tile(Db, d_rs, bias, M, Ncol, row0 + 16, col0 + j * 16, acc1[j], act, accum);
    }
}

// ------------------------ elementwise / reduction kernels ------------------------
__global__ void gw_zero(float* p, i64 n) {
    for (i64 i = blockIdx.x * (i64)blockDim.x + threadIdx.x; i < n; i += (i64)gridDim.x * blockDim.x)
        p[i] = 0.0f;
}

__global__ void gw_leaky(float* p, i64 n) {
    for (i64 i = blockIdx.x * (i64)blockDim.x + threadIdx.x; i < n; i += (i64)gridDim.x * blockDim.x) {
        float v = p[i];
        p[i] = v > 0.0f ? v : 0.01f * v;
    }
}

__global__ void gw_mul(const float* a, const float* b, float* o, i64 n) {
    for (i64 i = blockIdx.x * (i64)blockDim.x + threadIdx.x; i < n; i += (i64)gridDim.x * blockDim.x)
        o[i] = a[i] * b[i];
}

// global BN (no affine) stats for the single-channel input: out2 = {mean, rstd}
__global__ void gw_input_stats(const float* __restrict__ in, int n, float* __restrict__ out2) {
    __shared__ float ss[256], sq[256];
    float s = 0.0f, q = 0.0f;
    for (int i = threadIdx.x; i < n; i += 256) {
        float v = in[i];
        s += v; q += v * v;
    }
    ss[threadIdx.x] = s; sq[threadIdx.x] = q;
    __syncthreads();
    for (int w = 128; w > 0; w >>= 1) {
        if ((int)threadIdx.x < w) { ss[threadIdx.x] += ss[threadIdx.x + w]; sq[threadIdx.x] += sq[threadIdx.x + w]; }
        __syncthreads();
    }
    if (threadIdx.x == 0) {
        float m = ss[0] / n;
        float var = sq[0] / n - m * m;
        out2[0] = m;
        out2[1] = rsqrtf(var + EPS_);
    }
}

// x[b][c][t][n] = start_w[c]*norm(in[b][n][t]) + start_b[c]
__global__ void gw_start(const float* __restrict__ in, const float* __restrict__ st,
                         const float* __restrict__ sw, const float* __restrict__ sb,
                         float* __restrict__ x) {
    i64 total = (i64)B_ * C_ * T0_ * N_;
    for (i64 i = blockIdx.x * (i64)blockDim.x + threadIdx.x; i < total; i += (i64)gridDim.x * blockDim.x) {
        int n = (int)(i % N_);
        int t = (int)((i / N_) % T0_);
        int c = (int)((i / ((i64)N_ * T0_)) % C_);
        int b = (int)(i / ((i64)N_ * T0_ * C_));
        float v = in[(i64)b * N_ * T0_ + (i64)n * T0_ + t];
        v = (v - st[0]) * st[1];
        x[i] = sw[c] * v + sb[c];
    }
}

// adp = softmax(relu(nv1@nv2), axis=1) : one block per row, 256 threads
__global__ void gw_adp(const float* __restrict__ nv1, const float* __restrict__ nv2,
                       float* __restrict__ adp) {
    __shared__ float nvr[10];
    __shared__ float red[256];
    int row = blockIdx.x;
    if (threadIdx.x < 10) nvr[threadIdx.x] = nv1[row * 10 + threadIdx.x];
    __syncthreads();
    float vals[2];
    float mx = -1e30f;
#pragma unroll
    for (int j = 0; j < 2; ++j) {
        int m = threadIdx.x + j * 256;
        float v = -1e30f;
        if (m < N_) {
            float s = 0.0f;
#pragma unroll
            for (int k = 0; k < 10; ++k) s += nvr[k] * nv2[k * N_ + m];
            v = s > 0.0f ? s : 0.0f;
        }
        vals[j] = v;
        mx = fmaxf(mx, v);
    }
    red[threadIdx.x] = mx;
    __syncthreads();
    for (int w = 128; w > 0; w >>= 1) {
        if ((int)threadIdx.x < w) red[threadIdx.x] = fmaxf(red[threadIdx.x], red[threadIdx.x + w]);
        __syncthreads();
    }
    float rowmax = red[0];
    __syncthreads();
    float s = 0.0f;
#pragma unroll
    for (int j = 0; j < 2; ++j) {
        int m = threadIdx.x + j * 256;
        if (m < N_) { vals[j] = __expf(vals[j] - rowmax); s += vals[j]; }
    }
    red[threadIdx.x] = s;
    __syncthreads();
    for (int w = 128; w > 0; w >>= 1) {
        if ((int)threadIdx.x < w) red[threadIdx.x] += red[threadIdx.x + w];
        __syncthreads();
    }
    float tot = red[0];
#pragma unroll
    for (int j = 0; j < 2; ++j) {
        int m = threadIdx.x + j * 256;
        if (m < N_) adp[row * N_ + m] = vals[j] / tot;
    }
}

// G1w{f,b}[k][n] = w_edge_{f,b}[k] * G1[k][n]
__global__ void gw_scale_g1(const float* __restrict__ G1, const float* __restrict__ wf,
                            const float* __restrict__ wb, float* __restrict__ g1f,
                            float* __restrict__ g1b) {
    i64 total = (i64)EH_ * E_;
    for (i64 i = blockIdx.x * (i64)blockDim.x + threadIdx.x; i < total; i += (i64)gridDim.x * blockDim.x) {
        int k = (int)(i / E_);
        float v = G1[i];
        g1f[i] = wf[k] * v;
        g1b[i] = wb[k] * v;
    }
}

// xm[b][c][n] = mean_t x[b][c][t][n]
__global__ void gw_xmean(const float* __restrict__ x, float* __restrict__ xm, int Tcur) {
    i64 total = (i64)B_ * C_ * N_;
    for (i64 i = blockIdx.x * (i64)blockDim.x + threadIdx.x; i < total; i += (i64)gridDim.x * blockDim.x) {
        int n = (int)(i % N_);
        int c = (int)((i / N_) % C_);
        int b = (int)(i / ((i64)N_ * C_));
        const float* p = x + ((i64)(b * C_ + c) * Tcur) * N_ + n;
        float s = 0.0f;
        for (int t = 0; t < Tcur; ++t) s += p[(i64)t * N_];
        xm[i] = s / (float)Tcur;
    }
}

// edge scores: sc_{f,b}[b][e] = <src, vsrc> + <dst, vdst>
__global__ void gw_sc(const float* __restrict__ xm, const int* __restrict__ idx,
                      const float* __restrict__ vsf, const float* __restrict__ vdf,
                      const float* __restrict__ vsb, const float* __restrict__ vdb,
                      float* __restrict__ scf, float* __restrict__ scb) {
    i64 total = (i64)B_ * E_;
    for (i64 i = blockIdx.x * (i64)blockDim.x + threadIdx.x; i < total; i += (i64)gridDim.x * blockDim.x) {
        int e = (int)(i % E_);
        int b = (int)(i / E_);
        int i0 = idx[e];
        int i1 = idx[E_ + e];
        float sf = 0.0f, sb2 = 0.0f;
#pragma unroll
        for (int c = 0; c < C_; ++c) {
            float s = xm[(i64)(b * C_ + c) * N_ + i0];
            float d = xm[(i64)(b * C_ + c) * N_ + i1];
            sf  += s * vsf[c] + d * vdf[c];
            sb2 += s * vsb[c] + d * vdb[c];
        }
        scf[i] = sf;
        scb[i] = sb2;
    }
}

// s0[b][n][m] = (n==m)?A_fwd[n][m]:0 ; s1 likewise with A_bwd
__global__ void gw_sinit(const float* __restrict__ Af, const float* __restrict__ Abw,
                         float* __restrict__ s0, float* __restrict__ s1) {
    i64 total = (i64)B_ * N_ * N_;
    for (i64 i = blockIdx.x * (i64)blockDim.x + threadIdx.x; i < total; i += (i64)gridDim.x * blockDim.x) {
        int m = (int)(i % N_);
        int n = (int)((i / N_) % N_);
        s0[i] = (n == m) ? Af[(i64)n * N_ + m]  : 0.0f;
        s1[i] = (n == m) ? Abw[(i64)n * N_ + m] : 0.0f;
    }
}

// scatter gated edges: s0[b,i0,i1]=ef*A_fwd[i0,i1]; s1[b,i1,i0]=eb*A_bwd[i1,i0]
__global__ void gw_scatter(const float* __restrict__ ef, const float* __restrict__ eb,
                           const int* __restrict__ idx, const float* __restrict__ Af,
                           const float* __restrict__ Abw, float* __restrict__ s0,
                           float* __restrict__ s1) {
    i64 total = (i64)B_ * E_;
    for (i64 i = blockIdx.x * (i64)blockDim.x + threadIdx.x; i < total; i += (i64)gridDim.x * blockDim.x) {
        int e = (int)(i % E_);
        int b = (int)(i / E_);
        int i0 = idx[e];
        int i1 = idx[E_ + e];
        s0[(i64)b * N_ * N_ + (i64)i0 * N_ + i1] = ef[i] * Af[(i64)i0 * N_ + i1];
        s1[(i64)b * N_ * N_ + (i64)i1 * N_ + i0] = eb[i] * Abw[(i64)i1 * N_ + i0];
    }
}

// per-channel BN stats over (b, t, n); one block per channel
__global__ void gw_bn_stats(const float* __restrict__ x, int Tcur,
                            float* __restrict__ mean, float* __restrict__ rstd) {
    __shared__ float ss[256], sq[256];
    int c = blockIdx.x;
    int per = Tcur * N_;
    int cnt = B_ * per;
    float s = 0.0f, q = 0.0f;
    for (int i = threadIdx.x; i < cnt; i += 256) {
        int b = i / per;
        int r = i % per;
        float v = x[((i64)(b * C_ + c) * Tcur) * N_ + r];
        s += v; q += v * v;
    }
    ss[threadIdx.x] = s; sq[threadIdx.x] = q;
    __syncthreads();
    for (int w = 128; w > 0; w >>= 1) {
        if ((int)threadIdx.x < w) { ss[threadIdx.x] += ss[threadIdx.x + w]; sq[threadIdx.x] += sq[threadIdx.x + w]; }
        __syncthreads();
    }
    if (threadIdx.x == 0) {
        float m = ss[0] / cnt;
        float var = sq[0] / cnt - m * m;
        mean[c] = m;
        rstd[c] = rsqrtf(var + EPS_);
    }
}

__global__ void gw_bn_apply(float* __restrict__ x, const float* __restrict__ mean,
                            const float* __restrict__ rstd, const float* __restrict__ g,
                            const float* __restrict__ be, int Tcur) {
    i64 total = (i64)B_ * C_ * Tcur * N_;
    i64 per = (i64)Tcur * N_;
    for (i64 i = blockIdx.x * (i64)blockDim.x + threadIdx.x; i < total; i += (i64)gridDim.x * blockDim.x) {
        int c = (int)((i / per) % C_);
        x[i] = g[c] * (x[i] - mean[c]) * rstd[c] + be[c];
    }
}

// nxt[b,c,t,n] += cur[b,c,t+2,n]
__global__ void gw_resid(float* __restrict__ nxt, const float* __restrict__ cur,
                         int Tin, int Tout) {
    i64 total = (i64)B_ * C_ * Tout * N_;
    for (i64 i = blockIdx.x * (i64)blockDim.x + threadIdx.x; i < total; i += (i64)gridDim.x * blockDim.x) {
        int n = (int)(i % N_);
        int t = (int)((i / N_) % Tout);
        int c = (int)((i / ((i64)N_ * Tout)) % C_);
        int b = (int)(i / ((i64)N_ * Tout * C_));
        nxt[i] += cur[((i64)(b * C_ + c) * Tin + t + 2) * N_ + n];
    }
}

// [12][7][500] -> [12][500][7]
__global__ void gw_outtr(const float* __restrict__ src, float* __restrict__ dst) {
    i64 total = 12 * 7 * N_;
    for (i64 i = blockIdx.x * (i64)blockDim.x + threadIdx.x; i < total; i += (i64)gridDim.x * blockDim.x) {
        int n = (int)(i % N_);
        int t = (int)((i / N_) % 7);
        int o = (int)(i / (7 * N_));
        dst[(i64)o * N_ * 7 + (i64)n * 7 + t] = src[i];
    }
}

// ------------------------------ host side ------------------------------
static inline void launch_gemm(hipStream_t s,
                               const float* A, int ars, int acs, i64 abs_,
                               const float* B, int brs, i64 bbs, int baux, int bmode,
                               float* D, int drs, i64 dbs,
                               const float* bias, int M, int Ncol, int K, int nb, int epi) {
    dim3 blk(32, 4, 1);
    dim3 grd((Ncol + 255) / 256, (M + 31) / 32, nb);
    if (bmode == 0)
        gwnet_gemm_wmma<0><<<grd, blk, 0, s>>>(A, ars, acs, abs_, B, brs, bbs, baux,
                                               D, drs, dbs, bias, M, Ncol, K, epi);
    else
        gwnet_gemm_wmma<1><<<grd, blk, 0, s>>>(A, ars, acs, abs_, B, brs, bbs, baux,
                                               D, drs, dbs, bias, M, Ncol, K, epi);
}

static inline dim3 ewgrid(i64 total) {
    i64 g = (total + 255) / 256;
    if (g > 262144) g = 262144;
    return dim3((unsigned)g, 1, 1);
}

extern "C" void kernel_launch(void* const* d_in, const int* in_sizes, int n_in,
                              void* d_out, int out_size, void* d_ws, size_t ws_size,
                              hipStream_t stream) {
    (void)in_sizes; (void)n_in; (void)out_size; (void)ws_size;
    const float* input    = (const float*)d_in[0];
    const float* A_fwd    = (const float*)d_in[1];
    const float* A_bwd    = (const float*)d_in[2];
    const int*   indices  = (const int*)d_in[3];
    const float* G0_all   = (const float*)d_in[4];
    const float* G1_all   = (const float*)d_in[5];
    const float* nodevec1 = (const float*)d_in[6];
    const float* nodevec2 = (const float*)d_in[7];
    const float* w_edge_f = (const float*)d_in[8];
    const float* w_edge_b = (const float*)d_in[9];
    const float* vsrc_f   = (const float*)d_in[10];
    const float* vdst_f   = (const float*)d_in[11];
    const float* vsrc_b   = (const float*)d_in[12];
    const float* vdst_b   = (const float*)d_in[13];
    const float* start_w  = (const float*)d_in[14];
    const float* start_b  = (const float*)d_in[15];
    const float* filter_w = (const float*)d_in[16];
    const float* filter_b = (const float*)d_in[17];
    const float* gate_w   = (const float*)d_in[18];
    const float* gate_b   = (const float*)d_in[19];
    const float* gconv_w  = (const float*)d_in[20];
    const float* gconv_b  = (const float*)d_in[21];
    const float* bn_g_g   = (const float*)d_in[22];
    const float* bn_g_b   = (const float*)d_in[23];
    const float* bn_g     = (const float*)d_in[24];
    const float* bn_b     = (const float*)d_in[25];
    const float* skip_w   = (const float*)d_in[26];
    const float* skip_b   = (const float*)d_in[27];
    const float* end1_w   = (const float*)d_in[28];
    const float* end1_b   = (const float*)d_in[29];
    const float* end2_w   = (const float*)d_in[30];
    const float* end2_b   = (const float*)d_in[31];
    float* out = (float*)d_out;

    // -------- workspace carve-up --------
    float* w = (float*)d_ws;
    size_t off = 0;
    const size_t BIG = (size_t)B_ * C_ * T0_ * N_;   // 8.32M floats
    float* stats = w + off; off += 2;
    float* bufA  = w + off; off += BIG;
    float* bufB  = w + off; off += BIG;
    float* fgx   = w + off; off += BIG;
    float* t1    = w + off; off += BIG;
    float* t2    = w + off; off += BIG;
    float* adp   = w + off; off += (size_t)N_ * N_;
    float* g1f   = w + off; off += (size_t)EH_ * E_;
    float* g1b   = w + off; off += (size_t)EH_ * E_;
    float* GGf   = w + off; off += (size_t)E_ * E_;
    float* GGb   = w + off; off += (size_t)E_ * E_;
    float* s0    = w + off; off += (size_t)B_ * N_ * N_;
    float* s1    = w + off; off += (size_t)B_ * N_ * N_;
    float* xm    = w + off; off += (size_t)B_ * C_ * N_;
    float* scf   = w + off; off += (size_t)B_ * E_;
    float* scb   = w + off; off += (size_t)B_ * E_;
    float* efb   = w + off; off += (size_t)B_ * E_;
    float* ebb   = w + off; off += (size_t)B_ * E_;
    float* bnm   = w + off; off += C_;
    float* bnr   = w + off; off += C_;
    float* skip  = w + off; off += (size_t)B_ * 320 * 7 * N_;
    float* mid   = w + off; off += (size_t)640 * 7 * N_;
    float* otmp  = w + off; off += (size_t)12 * 7 * N_;

    // -------- setup stage --------
    gw_input_stats<<<1, 256, 0, stream>>>(input, B_ * N_ * T0_, stats);
    gw_start<<<ewgrid((i64)BIG), 256, 0, stream>>>(input, stats, start_w, start_b, bufA);
    gw_adp<<<N_, 256, 0, stream>>>(nodevec1, nodevec2, adp);
    gw_scale_g1<<<ewgrid((i64)EH_ * E_), 256, 0, stream>>>(G1_all, w_edge_f, w_edge_b, g1f, g1b);
    // GG = G0 @ (w*G1) : (2000x300)@(300x2000)
    launch_gemm(stream, G0_all, EH_, 1, 0, g1f, E_, 0, 0, 0, GGf, E_, 0, nullptr, E_, E_, EH_, 1, ACT_NONE);
    launch_gemm(stream, G0_all, EH_, 1, 0, g1b, E_, 0, 0, 0, GGb, E_, 0, nullptr, E_, E_, EH_, 1, ACT_NONE);
    gw_zero<<<ewgrid((i64)B_ * 320 * 7 * N_), 256, 0, stream>>>(skip, (i64)B_ * 320 * 7 * N_);

    // -------- 3 WaveNet blocks --------
    float* cur = bufA;
    float* nxt = bufB;
    int Tin = T0_;
    for (int blk = 0; blk < 3; ++blk) {
        int Tout = Tin - 2;
        int TinN  = Tin * N_;
        int ToutN = Tout * N_;
        i64 xbsIn  = (i64)C_ * TinN;
        i64 xbsOut = (i64)C_ * ToutN;

        // edge gating
        gw_xmean<<<ewgrid((i64)B_ * C_ * N_), 256, 0, stream>>>(cur, xm, Tin);
        gw_sc<<<ewgrid((i64)B_ * E_), 256, 0, stream>>>(xm, indices,
                vsrc_f + blk * C_, vdst_f + blk * C_, vsrc_b + blk * C_, vdst_b + blk * C_, scf, scb);
        // ef/eb = sigmoid(sc @ GG)   (32x2000)@(2000x2000)
        launch_gemm(stream, scf, E_, 1, 0, GGf, E_, 0, 0, 0, efb, E_, 0, nullptr, B_, E_, E_, 1, ACT_SIG);
        launch_gemm(stream, scb, E_, 1, 0, GGb, E_, 0, 0, 0, ebb, E_, 0, nullptr, B_, E_, E_, 1, ACT_SIG);
        gw_sinit<<<ewgrid((i64)B_ * N_ * N_), 256, 0, stream>>>(A_fwd, A_bwd, s0, s1);
        gw_scatter<<<ewgrid((i64)B_ * E_), 256, 0, stream>>>(efb, ebb, indices, A_fwd, A_bwd, s0, s1);

        // dilated convs (tanh filter, sigmoid gate), K' = 2*C, BMODE 1
        launch_gemm(stream, filter_w + (i64)blk * C_ * C_ * 2, 2 * C_, 1, 0,
                    cur, TinN, xbsIn, 2 * N_, 1,
                    t1, ToutN, xbsOut,
                    filter_b + blk * C_, C_, ToutN, 2 * C_, B_, ACT_TANH);
        launch_gemm(stream, gate_w + (i64)blk * C_ * C_ * 2, 2 * C_, 1, 0,
                    cur, TinN, xbsIn, 2 * N_, 1,
                    t2, ToutN, xbsOut,
                    gate_b + blk * C_, C_, ToutN, 2 * C_, B_, ACT_SIG);
        gw_mul<<<ewgrid((i64)B_ * xbsOut), 256, 0, stream>>>(t1, t2, fgx, (i64)B_ * xbsOut);

        // gconv projection part 0 (x itself) with bias
        const float* gwb = gconv_w + (i64)blk * 7 * C_ * C_;
        launch_gemm(stream, gwb, 1, C_, 0,
                    fgx, ToutN, xbsOut, 0, 0,
                    nxt, ToutN, xbsOut,
                    gconv_b + blk * C_, C_, ToutN, C_, B_, ACT_NONE);

        // graph supports: s0, s1 (batched), adp (broadcast)
        const float* sup[3] = { s0, s1, adp };
        i64 supbs[3] = { (i64)N_ * N_, (i64)N_ * N_, 0 };
        int p = 1;
        for (int si = 0; si < 3; ++si) {
            // x1 = x . A   (rows=(c,t), K=N, cols=N)
            launch_gemm(stream, fgx, N_, 1, xbsOut,
                        sup[si], N_, supbs[si], 0, 0,
                        t1, N_, xbsOut,
                        nullptr, C_ * Tout, N_, N_, B_, ACT_NONE);
            launch_gemm(stream, gwb + (i64)p * C_ * C_, 1, C_, 0,
                        t1, ToutN, xbsOut, 0, 0,
                        nxt, ToutN, xbsOut,
                        nullptr, C_, ToutN, C_, B_, ACT_ACC);
            ++p;
            // x2 = x1 . A
            launch_gemm(stream, t1, N_, 1, xbsOut,
                        sup[si], N_, supbs[si], 0, 0,
                        t2, N_, xbsOut,
                        nullptr, C_ * Tout, N_, N_, B_, ACT_NONE);
            launch_gemm(stream, gwb + (i64)p * C_ * C_, 1, C_, 0,
                        t2, ToutN, xbsOut, 0, 0,
                        nxt, ToutN, xbsOut,
                        nullptr, C_, ToutN, C_, B_, ACT_ACC);
            ++p;
        }

        // BN (affine) -> +residual -> BN (affine)
        gw_bn_stats<<<C_, 256, 0, stream>>>(nxt, Tout, bnm, bnr);
        gw_bn_apply<<<ewgrid((i64)B_ * xbsOut), 256, 0, stream>>>(nxt, bnm, bnr,
                bn_g_g + blk * C_, bn_g_b + blk * C_, Tout);
        gw_resid<<<ewgrid((i64)B_ * xbsOut), 256, 0, stream>>>(nxt, cur, Tin, Tout);
        gw_bn_stats<<<C_, 256, 0, stream>>>(nxt, Tout, bnm, bnr);
        gw_bn_apply<<<ewgrid((i64)B_ * xbsOut), 256, 0, stream>>>(nxt, bnm, bnr,
                bn_g + blk * C_, bn_b + blk * C_, Tout);

        // skip accumulation on trailing 7 timesteps: skip += skip_w @ x[..., -7:]
        launch_gemm(stream, skip_w + (i64)blk * 320 * C_, C_, 1, 0,
                    nxt + (i64)(Tout - 7) * N_, ToutN, xbsOut, 0, 0,
                    skip, 7 * N_, 320LL * 7 * N_,
                    skip_b + blk * 320, 320, 7 * N_, C_, B_, ACT_ACC);

        // ping-pong
        float* tmp = cur; cur = nxt; nxt = tmp;
        Tin = Tout;
    }

    // -------- head: leaky -> end1(leaky) -> end2 -> transpose-out --------
    gw_leaky<<<ewgrid((i64)B_ * 320 * 7 * N_), 256, 0, stream>>>(skip, (i64)B_ * 320 * 7 * N_);
    for (int b = 0; b < B_; ++b) {
        const float* skb = skip + (i64)b * 320 * 7 * N_;
        launch_gemm(stream, end1_w, 320, 1, 0,
                    skb, 7 * N_, 0, 0, 0,
                    mid, 7 * N_, 0,
                    end1_b, 640, 7 * N_, 320, 1, ACT_LEAKY);
        launch_gemm(stream, end2_w, 640, 1, 0,
                    mid, 7 * N_, 0, 0, 0,
                    otmp, 7 * N_, 0,
                    end2_b, 12, 7 * N_, 640, 1, ACT_NONE);
        gw_outtr<<<ewgrid(12LL * 7 * N_), 256, 0, stream>>>(otmp, out + (i64)b * 12 * N_ * 7);
    }
}